// SinkhornEvalAbs_14654428414011
// MI455X (gfx1250) — compile-verified
//
#include <hip/hip_runtime.h>
#include <hip/hip_bf16.h>

// ---------------------------------------------------------------------------
// Debiased Sinkhorn divergence on quaternion clouds, MI455X (gfx1250).
//  - Cost matrix via V_WMMA_F32_16X16X4_F32 (K=4 == quaternion dim, one WMMA
//    per 16x16 tile), acos applied once, S = -C/eps cached in workspace.
//  - 20 Sinkhorn iterations = 40 streaming LSE passes over the L2-resident
//    128MB S slab (16MB/batch << 192MB L2).
// ---------------------------------------------------------------------------

#define P     2048
#define NB    8
#define EPSV  0.05f
#define INVE  20.0f       // 1/eps

typedef __attribute__((ext_vector_type(2))) float v2f;
typedef __attribute__((ext_vector_type(8))) float v8f;

// ------------------------------- init --------------------------------------
__global__ __launch_bounds__(256) void k_logs(const float* __restrict__ wx,
                                              const float* __restrict__ wy,
                                              float* __restrict__ Lx,
                                              float* __restrict__ Ly) {
  int idx = blockIdx.x * 256 + threadIdx.x;
  if (idx < NB * P) {
    Lx[idx] = __logf(wx[idx]);
    Ly[idx] = __logf(wy[idx]);
  }
}

__global__ __launch_bounds__(256) void k_zero_fg(float* __restrict__ f,
                                                 float* __restrict__ g) {
  int idx = blockIdx.x * 256 + threadIdx.x;
  if (idx < NB * P) { f[idx] = 0.0f; g[idx] = 0.0f; }
}

// --------------------------- cost matrix (WMMA) ----------------------------
// S[n][i][j] = -(2/eps)*acos(min(|<x_i, y_j>|, 1-1e-7))
// One wave per 16x16 tile; A(16x4) and B(4x16) each need one float2 per lane:
//   lane L<16  -> (k0,k1) of point (tile*16 + L)
//   lane L>=16 -> (k2,k3) of point (tile*16 + L-16)
__global__ __launch_bounds__(256) void k_cost_wmma(const float* __restrict__ xr,
                                                   const float* __restrict__ yc,
                                                   float* __restrict__ S) {
  int wave = threadIdx.x >> 5;
  int lane = threadIdx.x & 31;
  int t    = blockIdx.x * 8 + wave;      // global tile id, grid is exact
  int n    = t >> 14;                    // / (128*128)
  int rem  = t & 16383;
  int ti   = rem >> 7;
  int tj   = rem & 127;
  int r    = lane & 15;
  int half = lane >> 4;

  const float* ap = xr + ((size_t)(n * P + ti * 16 + r)) * 4 + half * 2;
  const float* bp = yc + ((size_t)(n * P + tj * 16 + r)) * 4 + half * 2;
  v2f a; a[0] = ap[0]; a[1] = ap[1];
  v2f b; b[0] = bp[0]; b[1] = bp[1];
  v8f c = {};
  // D = A * B  (C = 0):  v_wmma_f32_16x16x4_f32
  c = __builtin_amdgcn_wmma_f32_16x16x4_f32(false, a, false, b,
                                            (short)0, c, false, false);

  // C/D layout: VGPR q, lane L: N = L&15, M = q + (L>=16 ? 8 : 0)
  int j     = tj * 16 + r;
  int mbase = ti * 16 + half * 8;
  size_t base = (size_t)n * P * P;
#pragma unroll
  for (int q = 0; q < 8; ++q) {
    float inner = fminf(fabsf(c[q]), 1.0f - 1e-7f);
    float cost  = 2.0f * acosf(inner);
    S[base + (size_t)(mbase + q) * P + j] = -cost * INVE;
  }
}

// ------------------- g-update: column LSE, chunked -------------------------
// grid.x = n(8) * jt(8) * rc(8) = 512, block = 256 threads (one column each).
__global__ __launch_bounds__(256) void k_gpart(const float* __restrict__ S,
                                               const float* __restrict__ La,
                                               const float* __restrict__ f,
                                               float2* __restrict__ part) {
  int bid = blockIdx.x;
  int n  = bid >> 6;
  int jt = (bid >> 3) & 7;
  int rc = bid & 7;
  int tid = threadIdx.x;

  __shared__ float qsh[256];
  int i0 = rc * 256;
  int gi = n * P + i0 + tid;
  qsh[tid] = La[gi] + f[gi] * INVE;     // la_i + f_i/eps for this row chunk
  __syncthreads();

  int j = jt * 256 + tid;
  const float* Scol = S + (size_t)n * P * P + (size_t)i0 * P + j;
  float m = -INFINITY, s = 0.0f;
  for (int ii = 0; ii < 256; ++ii) {
    float v = qsh[ii] + Scol[(size_t)ii * P];
    if (v > m) { s = s * __expf(m - v) + 1.0f; m = v; }
    else       { s += __expf(v - m); }
  }
  part[((size_t)(n * P + j)) * 8 + rc] = make_float2(m, s);
}

__global__ __launch_bounds__(256) void k_gcomb(const float2* __restrict__ part,
                                               float* __restrict__ g) {
  int idx = blockIdx.x * 256 + threadIdx.x;   // over NB*P columns
  if (idx >= NB * P) return;
  const float2* pp = part + (size_t)idx * 8;
  float m = -INFINITY;
#pragma unroll
  for (int c = 0; c < 8; ++c) m = fmaxf(m, pp[c].x);
  float s = 0.0f;
#pragma unroll
  for (int c = 0; c < 8; ++c) s += pp[c].y * __expf(pp[c].x - m);
  g[idx] = -EPSV * (m + __logf(s));
}

// ---------------------- f-update: row LSE, one block/row -------------------
__global__ __launch_bounds__(256) void k_frow(const float* __restrict__ S,
                                              const float* __restrict__ Lb,
                                              const float* __restrict__ g,
                                              float* __restrict__ f) {
  int bid = blockIdx.x;                 // n*P + i
  int n = bid >> 11;
  int i = bid & (P - 1);
  int tid = threadIdx.x;

  const float* Srow = S + (size_t)n * P * P + (size_t)i * P;
  const float* Lbr  = Lb + n * P;
  const float* gr   = g  + n * P;

  float m = -INFINITY, s = 0.0f;
  for (int j = tid; j < P; j += 256) {
    float v = Lbr[j] + gr[j] * INVE + Srow[j];
    if (v > m) { s = s * __expf(m - v) + 1.0f; m = v; }
    else       { s += __expf(v - m); }
  }
  // wave32 reduction of (m, s)
#pragma unroll
  for (int off = 16; off; off >>= 1) {
    float mo = __shfl_xor(m, off);
    float so = __shfl_xor(s, off);
    float M  = fmaxf(m, mo);
    s = s * __expf(m - M) + so * __expf(mo - M);
    m = M;
  }
  __shared__ float2 wr[8];
  if ((tid & 31) == 0) wr[tid >> 5] = make_float2(m, s);
  __syncthreads();
  if (tid == 0) {
    float M = wr[0].x, Sv = wr[0].y;
#pragma unroll
    for (int w = 1; w < 8; ++w) {
      float mo = wr[w].x, so = wr[w].y;
      float M2 = fmaxf(M, mo);
      Sv = Sv * __expf(M - M2) + so * __expf(mo - M2);
      M = M2;
    }
    f[bid] = -EPSV * (M + __logf(Sv));
  }
}

// ----------------------------- epilogue ------------------------------------
__global__ __launch_bounds__(256) void k_dot(const float* __restrict__ a,
                                             const float* __restrict__ b,
                                             const float* __restrict__ f,
                                             const float* __restrict__ g,
                                             float* __restrict__ res) {
  int n = blockIdx.x;
  int tid = threadIdx.x;
  float acc = 0.0f;
  for (int i = tid; i < P; i += 256)
    acc += a[n * P + i] * f[n * P + i] + b[n * P + i] * g[n * P + i];
#pragma unroll
  for (int off = 16; off; off >>= 1) acc += __shfl_xor(acc, off);
  __shared__ float wr[8];
  if ((tid & 31) == 0) wr[tid >> 5] = acc;
  __syncthreads();
  if (tid == 0) {
    float t = 0.0f;
#pragma unroll
    for (int w = 0; w < 8; ++w) t += wr[w];
    res[n] = t;
  }
}

__global__ void k_final(const float* __restrict__ res, float* __restrict__ out) {
  if (threadIdx.x == 0 && blockIdx.x == 0) {
    float acc = 0.0f;
    for (int n = 0; n < NB; ++n)
      acc += res[n] - 0.5f * (res[NB + n] + res[2 * NB + n]);
    out[0] = acc * (1.0f / (float)NB);
  }
}

// ---------------------------------------------------------------------------
extern "C" void kernel_launch(void* const* d_in, const int* in_sizes, int n_in,
                              void* d_out, int out_size, void* d_ws, size_t ws_size,
                              hipStream_t stream) {
  // setup_inputs order: y, w_y, x, w_x
  const float* y  = (const float*)d_in[0];
  const float* wy = (const float*)d_in[1];
  const float* x  = (const float*)d_in[2];
  const float* wx = (const float*)d_in[3];
  float* out = (float*)d_out;

  // workspace carve-up
  uint8_t* w = (uint8_t*)d_ws;
  float*  S    = (float*)w;  w += (size_t)NB * P * P * sizeof(float);   // 128 MB
  float*  Lx   = (float*)w;  w += (size_t)NB * P * sizeof(float);
  float*  Ly   = (float*)w;  w += (size_t)NB * P * sizeof(float);
  float*  f    = (float*)w;  w += (size_t)NB * P * sizeof(float);
  float*  g    = (float*)w;  w += (size_t)NB * P * sizeof(float);
  float2* part = (float2*)w; w += (size_t)NB * P * 8 * sizeof(float2);  // 1 MB
  float*  res  = (float*)w;  w += 3 * NB * sizeof(float);

  k_logs<<<64, 256, 0, stream>>>(wx, wy, Lx, Ly);

  // problems: 0 = OT(x,y), 1 = OT(x,x), 2 = OT(y,y)
  const float* rows[3] = { x,  x,  y  };
  const float* cols[3] = { y,  x,  y  };
  const float* Las[3]  = { Lx, Lx, Ly };
  const float* Lbs[3]  = { Ly, Lx, Ly };
  const float* aws[3]  = { wx, wx, wy };
  const float* bws[3]  = { wy, wx, wy };

  for (int p = 0; p < 3; ++p) {
    k_zero_fg<<<64, 256, 0, stream>>>(f, g);
    k_cost_wmma<<<16384, 256, 0, stream>>>(rows[p], cols[p], S);
    for (int it = 0; it < 20; ++it) {
      k_gpart<<<512,   256, 0, stream>>>(S, Las[p], f, part);
      k_gcomb<<<64,    256, 0, stream>>>(part, g);
      k_frow <<<16384, 256, 0, stream>>>(S, Lbs[p], g, f);
    }
    k_dot<<<NB, 256, 0, stream>>>(aws[p], bws[p], f, g, res + p * NB);
  }
  k_final<<<1, 1, 0, stream>>>(res, out);
}